// MLA_18519898980689
// MI455X (gfx1250) — compile-verified
//
#include <hip/hip_runtime.h>

// MLA decode, weight-absorbed, fused cache-streaming.
// H=16 ND=128 RD=64 VD=128 KVR=512, B=32, DIM=2048, kv_len=4096.
// All GEMMs via V_WMMA_F32_16X16X4_F32 (fp32, matches reference precision).

typedef float v2f __attribute__((ext_vector_type(2)));
typedef float v8f __attribute__((ext_vector_type(8)));

// ---------------------------------------------------------------------------
// Generic fp32 WMMA GEMM: C[M,N] = scale * A[M,K] @ B^T  (+ C if ACC)
//   BT=true : B stored [N,K], element(n,k) = B[n*ldb + k]   (weight W[out,in])
//   BT=false: B stored [K,N], element(n,k) = B[k*ldb + n]
// One wave per 16x16 tile. grid = (N/16, M/16, Z). M,N %16==0, K %4==0.
// ---------------------------------------------------------------------------
template <bool BT, bool ACC>
__global__ void wmma_gemm_f32(const float* __restrict__ A,
                              const float* __restrict__ B,
                              float* __restrict__ C,
                              int K,
                              long lda, long ldb, long ldc,
                              long Az, long Bz, long Cz,
                              float scale)
{
    const int lane = threadIdx.x;       // blockDim.x == 32 (one wave)
    const int idx  = lane & 15;
    const int half = lane >> 4;         // 0 or 1
    const int n0 = blockIdx.x * 16;
    const int m0 = blockIdx.y * 16;

    A += (long)blockIdx.z * Az;
    B += (long)blockIdx.z * Bz;
    C += (long)blockIdx.z * Cz;

    const float* __restrict__ arow = A + (long)(m0 + idx) * lda + 2 * half;
    const float* __restrict__ brow;
    if (BT) brow = B + (long)(n0 + idx) * ldb + 2 * half;
    else    brow = B + (long)(2 * half) * ldb + (n0 + idx);

    v8f acc = {};
    for (int k = 0; k < K; k += 4) {
        v2f a, b;
        a.x = arow[k];
        a.y = arow[k + 1];
        if (BT) {
            b.x = brow[k];
            b.y = brow[k + 1];
        } else {
            b.x = brow[(long)k * ldb];
            b.y = brow[(long)(k + 1) * ldb];
        }
        acc = __builtin_amdgcn_wmma_f32_16x16x4_f32(
            false, a, false, b, (short)0, acc, false, false);
    }

    #pragma unroll
    for (int j = 0; j < 8; ++j) {
        long r = (long)(m0 + j + 8 * half) * ldc + (n0 + idx);
        float v = acc[j] * scale;
        if (ACC) v += C[r];
        C[r] = v;
    }
}

// ---------------------------------------------------------------------------
// Fused cache streamer: per block, one (batch, 128-kv-row chunk):
//   1) stage chunk [128 x 576] from cache_prefix through LDS (row stride 580
//      floats: 16B-aligned rows, conflict-free ds_load_b64) and write the
//      kv_lora output in the same pass (row 4095 is re-read from kvl, where
//      the c_kv_new GEMM already put it).
//   2) 8 waves x one 16-kv score tile each: scores[16h x 16kv] =
//      Qeff(absorbed,pre-scaled) @ Keff^T via WMMA, with the 64 RoPE channels
//      rotated on the fly from LDS.
// Single pass over the cache: copy + rope + score GEMM share one HBM read.
// ---------------------------------------------------------------------------
#define CHUNK 128
#define TSTR  580                      // LDS row stride (floats)

__global__ void __launch_bounds__(256)
fused_stream_score_kernel(const float* __restrict__ pre,
                          float* __restrict__ kvl,
                          const float* __restrict__ Qeff,
                          float* __restrict__ scores)
{
    extern __shared__ float smem[];    // CHUNK * TSTR floats (~290 KB)
    const int t     = threadIdx.x;
    const int chunk = blockIdx.x;      // 0..31
    const int b     = blockIdx.y;      // 0..31
    const long p0   = (long)chunk * CHUNK;

    // ---- stage + copy: 128 rows x 144 float4 = 18432 float4, 72/thread ----
    #pragma unroll 2
    for (int it = 0; it < 72; ++it) {
        int idx4 = t + it * 256;
        int r  = idx4 / 144;
        int c4 = idx4 - r * 144;
        long p = p0 + r;
        const float4* src = (p < 4095)
            ? (const float4*)(pre + ((long)b * 4095 + p) * 576) + c4
            : (const float4*)(kvl + ((long)b * 4096 + p) * 576) + c4;
        if (it + 1 < 72) __builtin_prefetch(src + 256, 0, 0);  // global_prefetch
        float4 v = *src;
        *(float4*)(smem + r * TSTR + c4 * 4) = v;
        ((float4*)(kvl + ((long)b * 4096 + p) * 576))[c4] = v; // kv_lora out
    }
    __syncthreads();

    // ---- scores: wave w handles local kv rows [w*16, w*16+16) ----
    const int lane = t & 31;
    const int wave = t >> 5;
    const int idx  = lane & 15;
    const int half = lane >> 4;
    const int r0   = wave * 16;

    const float* __restrict__ arow = Qeff + ((long)b * 16 + idx) * 576 + 2 * half;
    const float* __restrict__ brow = smem + (r0 + idx) * TSTR + 2 * half;

    v8f acc = {};
    // nope part: K = 0..511 straight from LDS
    for (int k = 0; k < 512; k += 4) {
        v2f a, bb;
        a.x  = arow[k];     a.y  = arow[k + 1];
        bb.x = brow[k];     bb.y = brow[k + 1];
        acc = __builtin_amdgcn_wmma_f32_16x16x4_f32(
            false, a, false, bb, (short)0, acc, false, false);
    }
    // rope part: K = 512..575, rotate k_rope(pos p) on the fly from LDS
    {
        const long  p = p0 + r0 + idx;                 // kv position of my row
        const float* __restrict__ T = smem + (r0 + idx) * TSTR + 512;
        for (int k = 0; k < 64; k += 4) {
            int i0 = k + 2 * half;                     // even: pair stays in half
            v2f a, bb;
            a.x = arow[512 + k];
            a.y = arow[512 + k + 1];
            {
                int j = i0 & 31;
                float th = __powf(10000.0f, -(float)j * (1.0f / 32.0f));
                float s, c; __sincosf((float)p * th, &s, &c);
                float x1 = T[j], x2 = T[32 + j];
                bb.x = (i0 < 32) ? (x1 * c - x2 * s) : (x2 * c + x1 * s);
            }
            {
                int j = (i0 + 1) & 31;
                float th = __powf(10000.0f, -(float)j * (1.0f / 32.0f));
                float s, c; __sincosf((float)p * th, &s, &c);
                float x1 = T[j], x2 = T[32 + j];
                bb.y = (i0 + 1 < 32) ? (x1 * c - x2 * s) : (x2 * c + x1 * s);
            }
            acc = __builtin_amdgcn_wmma_f32_16x16x4_f32(
                false, a, false, bb, (short)0, acc, false, false);
        }
    }
    // store: scores[b][h = j+8*half][p0 + r0 + idx]   (Qeff pre-scaled)
    #pragma unroll
    for (int j = 0; j < 8; ++j)
        scores[((long)b * 16 + j + 8 * half) * 4096 + p0 + r0 + idx] = acc[j];
}

// ---------------------------------------------------------------------------
// RoPE the query rope-part (pos 4095), scaled, into Qeff[...,512:576].
// ---------------------------------------------------------------------------
__global__ void q_rope_kernel(const float* __restrict__ qnr,
                              float* __restrict__ Qeff, float scale)
{
    int t  = threadIdx.x;
    int lr = t >> 5;
    int i  = t & 31;
    int row = blockIdx.x * 4 + lr;                 // b*16 + h
    if (row >= 512) return;
    int b = row >> 4, h = row & 15;
    const float* x = qnr + (long)b * 3072 + h * 192 + 128;
    float theta = __powf(10000.0f, -(float)i * (1.0f / 32.0f));
    float s, c;
    __sincosf(4095.0f * theta, &s, &c);
    float* o = Qeff + (long)row * 576 + 512;
    o[i]      = (x[i] * c - x[i + 32] * s) * scale;
    o[i + 32] = (x[i + 32] * c + x[i] * s) * scale;
}

// ---------------------------------------------------------------------------
// Row softmax over scores [512 rows, 4096 cols], in place (logits pre-scaled).
// ---------------------------------------------------------------------------
__global__ void softmax_kernel(float* __restrict__ scores)
{
    __shared__ float redm[8];
    __shared__ float reds[8];
    const int t = threadIdx.x;                     // 256 threads = 8 waves
    float* p = scores + (long)blockIdx.x * 4096;

    float mx = -1e30f;
    for (int i = t; i < 4096; i += 256) mx = fmaxf(mx, p[i]);
    for (int o = 16; o; o >>= 1) mx = fmaxf(mx, __shfl_xor(mx, o, 32));
    if ((t & 31) == 0) redm[t >> 5] = mx;
    __syncthreads();
    if (t < 8) {
        float m = redm[t];
        for (int o = 4; o; o >>= 1) m = fmaxf(m, __shfl_xor(m, o, 32));
        if (t == 0) redm[0] = m;
    }
    __syncthreads();
    mx = redm[0];

    float sum = 0.0f;
    for (int i = t; i < 4096; i += 256) {
        float e = __expf(p[i] - mx);
        p[i] = e;
        sum += e;
    }
    for (int o = 16; o; o >>= 1) sum += __shfl_xor(sum, o, 32);
    if ((t & 31) == 0) reds[t >> 5] = sum;
    __syncthreads();
    if (t < 8) {
        float s2 = reds[t];
        for (int o = 4; o; o >>= 1) s2 += __shfl_xor(s2, o, 32);
        if (t == 0) reds[0] = s2;
    }
    __syncthreads();
    float inv = 1.0f / reds[0];
    for (int i = t; i < 4096; i += 256) p[i] *= inv;
}

// ---------------------------------------------------------------------------
extern "C" void kernel_launch(void* const* d_in, const int* in_sizes, int n_in,
                              void* d_out, int out_size, void* d_ws, size_t ws_size,
                              hipStream_t stream)
{
    const float* x    = (const float*)d_in[0];   // [32,1,2048]
    const float* pre  = (const float*)d_in[1];   // [32,4095,576]
    const float* Wqd  = (const float*)d_in[2];   // [768,2048]
    const float* Wqu  = (const float*)d_in[3];   // [3072,768]
    const float* Wkvd = (const float*)d_in[4];   // [576,2048]
    const float* Wkvu = (const float*)d_in[5];   // [4096,512]
    const float* Wo   = (const float*)d_in[6];   // [2048,2048]

    float* y   = (float*)d_out;                  // [32,2048]
    float* kvl = (float*)d_out + 32L * 2048;     // [32,4096,576]

    // workspace partition (floats)
    float* ws      = (float*)d_ws;
    float* q_lora  = ws;                         // 32*768
    float* qnr     = q_lora + 32L * 768;         // 32*3072
    float* Qeff    = qnr + 32L * 3072;           // 32*16*576 (scaled)
    float* scores  = Qeff + 32L * 16 * 576;      // 32*16*4096
    float* o_lora  = scores + 32L * 16 * 4096;   // 32*16*512
    float* o_flat  = o_lora + 32L * 16 * 512;    // 32*2048

    const float scale = 0.072168783649f;         // 1/sqrt(192)
    const dim3 wave(32);

    // 1) new-token latent: kvl[b,4095,:] = x @ Wkvd^T   (must precede fusion)
    wmma_gemm_f32<true, false><<<dim3(36, 2, 1), wave, 0, stream>>>(
        x, Wkvd, kvl + 4095L * 576, 2048, 2048, 2048, 4096L * 576, 0, 0, 0, 1.0f);

    // 2) q path: q_lora = x @ Wqd^T ; qnr = q_lora @ Wqu^T
    wmma_gemm_f32<true, false><<<dim3(48, 2, 1), wave, 0, stream>>>(
        x, Wqd, q_lora, 2048, 2048, 2048, 768, 0, 0, 0, 1.0f);
    wmma_gemm_f32<true, false><<<dim3(192, 2, 1), wave, 0, stream>>>(
        q_lora, Wqu, qnr, 768, 768, 768, 3072, 0, 0, 0, 1.0f);

    // 3) absorb k-up-proj into q: Qeff[b,h,0:512] = scale * q_nope @ Wk_h
    wmma_gemm_f32<false, false><<<dim3(32, 2, 16), wave, 0, stream>>>(
        qnr, Wkvu, Qeff, 128, 3072, 512, 16L * 576,
        /*Az=*/192, /*Bz=*/256L * 512, /*Cz=*/576, scale);

    // 4) Qeff[b,h,512:576] = scale * rope(q_rope, pos 4095)
    q_rope_kernel<<<128, 128, 0, stream>>>(qnr, Qeff, scale);

    // 5) fused: copy cache -> kv_lora, RoPE K, scores = Qeff @ Keff^T
    //    one HBM pass over the cache; 297KB dynamic LDS per workgroup
    fused_stream_score_kernel<<<dim3(32, 32), 256, CHUNK * TSTR * 4, stream>>>(
        pre, kvl, Qeff, scores);

    // 6) softmax over kv dimension
    softmax_kernel<<<512, 256, 0, stream>>>(scores);

    // 7) o_lora = attn @ kv_nope  (z = batch; B in [K=4096,N=512] layout)
    wmma_gemm_f32<false, false><<<dim3(32, 1, 32), wave, 0, stream>>>(
        scores, kvl, o_lora, 4096, 4096, 576, 512,
        16L * 4096, 4096L * 576, 16L * 512, 1.0f);

    // 8) per-head v projection: o_flat[:,h*128:+128] = o_lora[:,h,:] @ Wv_h^T
    wmma_gemm_f32<true, false><<<dim3(8, 2, 16), wave, 0, stream>>>(
        o_lora, Wkvu + 128L * 512, o_flat, 512, 16L * 512, 512, 2048,
        /*Az=*/512, /*Bz=*/256L * 512, /*Cz=*/128, 1.0f);

    // 9) y = o_flat @ Wo^T
    wmma_gemm_f32<true, false><<<dim3(128, 2, 1), wave, 0, stream>>>(
        o_flat, Wo, y, 2048, 2048, 2048, 2048, 0, 0, 0, 1.0f);
}